// CausalSelfAttentionWithCache_8650064134643
// MI455X (gfx1250) — compile-verified
//
#include <hip/hip_runtime.h>
#include <hip/hip_bf16.h>

#define BB   32
#define SS   2048
#define DD   128
#define HH   16
#define DT   (DD * HH)      // 2048
#define QKVN (3 * DT)       // 6144
#define SP1  2049
#define NCHUNK 8
#define CHUNK  256          // NCHUNK * CHUNK == SS
#define PSTRIDE 132         // per-(bh,chunk) partial: [0]=m, [1]=l, [4..131]=acc
#define EPS  1e-5f
#define SCALE 0.08838834764831845f   // 1/sqrt(128)

typedef __attribute__((ext_vector_type(2))) float v2f;
typedef __attribute__((ext_vector_type(4))) float v4f;
typedef __attribute__((ext_vector_type(8))) float v8f;

// ---------------------------------------------------------------------------
// Generic small-M f32 GEMM using V_WMMA_F32_16X16X4_F32.
// Computes Out[m][n] = sum_k A[m][k] * Wrow[n][k] + bias[n] (+ resid[m][n]).
// One wave per 16x16 output tile. ISA VGPR layouts (CDNA5 ISA):
//   A 16x4 f32 : lanes 0-15 -> K=0,1 in v[0],v[1]; lanes 16-31 -> K=2,3
//   B 4x16 f32 : v[0]: K=0 (lanes 0-15) / K=2 (lanes 16-31); v[1]: K=1 / K=3
//   C 16x16 f32: VGPR r: lanes 0-15 -> M=r, lanes 16-31 -> M=r+8
// ---------------------------------------------------------------------------
__global__ __launch_bounds__(32)
void wmma_gemm_bias(const float* __restrict__ A, int lda,
                    const float* __restrict__ Wrow,   // N x K, row-major
                    const float* __restrict__ bias,   // N
                    const float* __restrict__ resid,  // M x ldo or nullptr
                    float* __restrict__ Out, int ldo, int K)
{
    const int lane = threadIdx.x;
    const int half = lane >> 4;       // 0: lanes 0-15, 1: lanes 16-31
    const int l16  = lane & 15;
    const int m0   = blockIdx.y << 4;
    const int n0   = blockIdx.x << 4;

    const float* ap = A    + (size_t)(m0 + l16) * lda + (half << 1);
    const float* bp = Wrow + (size_t)(n0 + l16) * K   + (half << 1);

    v8f c = {0.f, 0.f, 0.f, 0.f, 0.f, 0.f, 0.f, 0.f};
#pragma unroll 4
    for (int k = 0; k < K; k += 4) {
        v2f a; a[0] = ap[k]; a[1] = ap[k + 1];
        v2f b; b[0] = bp[k]; b[1] = bp[k + 1];
        c = __builtin_amdgcn_wmma_f32_16x16x4_f32(
                /*neg_a=*/false, a, /*neg_b=*/false, b,
                /*c_mod=*/(short)0, c, /*reuse_a=*/false, /*reuse_b=*/false);
    }

    const int   col = n0 + l16;
    const float bv  = bias[col];
#pragma unroll
    for (int r = 0; r < 8; ++r) {
        const int row = m0 + r + (half << 3);
        float v = c[r] + bv;
        if (resid) v += resid[(size_t)row * ldo + col];
        Out[(size_t)row * ldo + col] = v;
    }
}

// ---------------------------------------------------------------------------
// Fused KV-cache copy + flash-decode partial attention.
// Grid: B*H*NCHUNK blocks, 256 threads (8 waves). Each wave owns 32 keys;
// one key row (128 f32) is covered by one wave (v4f per lane -> one fully
// coalesced 512B global_load_b128 / global_store_b128 per row).
// K and V are each read exactly once from HBM; the ~2.1 GB stream is tagged
// non-temporal (ISA TH=NT) so it does not thrash the 192 MB L2.
// ---------------------------------------------------------------------------
__global__ __launch_bounds__(256)
void attn_stream(const float* __restrict__ pk, const float* __restrict__ pv,
                 const float* __restrict__ qkv,
                 float* __restrict__ outK, float* __restrict__ outV,
                 float* __restrict__ part)
{
    const int bh    = blockIdx.x / NCHUNK;
    const int chunk = blockIdx.x - bh * NCHUNK;
    const int b = bh >> 4, h = bh & 15;
    const int tid  = threadIdx.x;
    const int wave = tid >> 5, lane = tid & 31;

    const v4f q4 = *(const v4f*)(qkv + (size_t)b * QKVN + h * DD + lane * 4);

    const size_t kvIn  = (size_t)bh * SS  * DD;
    const size_t kvOut = (size_t)bh * SP1 * DD;
    const int    j0    = chunk * CHUNK + wave * 32;

    __shared__ float sc[CHUNK];
    __shared__ float accW[8][DD];
    __shared__ float lW[8];
    __shared__ float red8[8];

    // ---- pass 1: scores + copy K rows to output cache (NT stream) ----
    for (int i = 0; i < 32; ++i) {
        const int j = j0 + i;
        const v4f kv = __builtin_nontemporal_load(
            (const v4f*)(pk + kvIn + (size_t)j * DD + lane * 4));
        __builtin_nontemporal_store(
            kv, (v4f*)(outK + kvOut + (size_t)j * DD + lane * 4));
        float p = kv.x * q4.x + kv.y * q4.y + kv.z * q4.z + kv.w * q4.w;
#pragma unroll
        for (int off = 16; off >= 1; off >>= 1) p += __shfl_xor(p, off, 32);
        if (lane == 0) sc[wave * 32 + i] = p * SCALE;
    }
    __syncthreads();

    // ---- block max over the 256 chunk scores ----
    float m = sc[tid];
#pragma unroll
    for (int off = 16; off >= 1; off >>= 1) m = fmaxf(m, __shfl_xor(m, off, 32));
    if (lane == 0) red8[wave] = m;
    __syncthreads();
    m = red8[0];
#pragma unroll
    for (int w = 1; w < 8; ++w) m = fmaxf(m, red8[w]);

    // ---- pass 2: exp-weighted V accumulation + copy V rows (NT stream) ----
    v4f  acc = {0.f, 0.f, 0.f, 0.f};
    float lsum = 0.f;
    for (int i = 0; i < 32; ++i) {
        const int   j = j0 + i;
        const float p = __expf(sc[wave * 32 + i] - m);
        const v4f vv = __builtin_nontemporal_load(
            (const v4f*)(pv + kvIn + (size_t)j * DD + lane * 4));
        __builtin_nontemporal_store(
            vv, (v4f*)(outV + kvOut + (size_t)j * DD + lane * 4));
        acc.x += p * vv.x; acc.y += p * vv.y; acc.z += p * vv.z; acc.w += p * vv.w;
        lsum  += p;
    }
    accW[wave][lane * 4 + 0] = acc.x;
    accW[wave][lane * 4 + 1] = acc.y;
    accW[wave][lane * 4 + 2] = acc.z;
    accW[wave][lane * 4 + 3] = acc.w;
    if (lane == 0) lW[wave] = lsum;
    __syncthreads();

    float* pp = part + (size_t)(bh * NCHUNK + chunk) * PSTRIDE;
    if (tid < DD) {
        float a = 0.f;
#pragma unroll
        for (int w = 0; w < 8; ++w) a += accW[w][tid];
        pp[4 + tid] = a;
    } else if (tid == DD) {
        float L = 0.f;
#pragma unroll
        for (int w = 0; w < 8; ++w) L += lW[w];
        pp[0] = m;
        pp[1] = L;
    }
}

// ---------------------------------------------------------------------------
// Per-(b,h): append new K/V row at s=2048, fold in its score, merge the 8
// chunk partials with log-sum-exp rescaling, emit attention output y.
// ---------------------------------------------------------------------------
__global__ __launch_bounds__(128)
void attn_reduce(const float* __restrict__ qkv, const float* __restrict__ part,
                 float* __restrict__ outK, float* __restrict__ outV,
                 float* __restrict__ y)
{
    const int bh = blockIdx.x, b = bh >> 4, h = bh & 15;
    const int d  = threadIdx.x;

    const float qd  = qkv[(size_t)b * QKVN + h * DD + d];
    const float knd = qkv[(size_t)b * QKVN + (HH + h) * DD + d];
    const float vnd = qkv[(size_t)b * QKVN + (2 * HH + h) * DD + d];

    outK[((size_t)bh * SP1 + SS) * DD + d] = knd;
    outV[((size_t)bh * SP1 + SS) * DD + d] = vnd;

    __shared__ float red[128];
    red[d] = qd * knd;
    __syncthreads();
    for (int s = 64; s > 0; s >>= 1) {
        if (d < s) red[d] += red[d + s];
        __syncthreads();
    }
    const float snew = red[0] * SCALE;

    const float* pp = part + (size_t)bh * NCHUNK * PSTRIDE;
    float M = snew;
#pragma unroll
    for (int c = 0; c < NCHUNK; ++c) M = fmaxf(M, pp[c * PSTRIDE]);

    const float wnew = __expf(snew - M);
    float L   = wnew;
    float num = wnew * vnd;
#pragma unroll
    for (int c = 0; c < NCHUNK; ++c) {
        const float w = __expf(pp[c * PSTRIDE] - M);
        L   += pp[c * PSTRIDE + 1] * w;
        num += pp[c * PSTRIDE + 4 + d] * w;
    }
    y[(size_t)b * DT + h * DD + d] = num / L;
}

// ---------------------------------------------------------------------------
// LayerNorm over the last dim (128), no affine.
// ---------------------------------------------------------------------------
__global__ __launch_bounds__(128)
void layernorm(const float* __restrict__ z, float* __restrict__ out)
{
    const int b = blockIdx.x, d = threadIdx.x;
    const float v = z[(size_t)b * DD + d];
    __shared__ float s1[128];
    __shared__ float s2[128];
    s1[d] = v;
    s2[d] = v * v;
    __syncthreads();
    for (int s = 64; s > 0; s >>= 1) {
        if (d < s) { s1[d] += s1[d + s]; s2[d] += s2[d + s]; }
        __syncthreads();
    }
    const float mu  = s1[0] * (1.f / DD);
    const float var = s2[0] * (1.f / DD) - mu * mu;
    out[(size_t)b * DD + d] = (v - mu) * rsqrtf(var + EPS);
}

// ---------------------------------------------------------------------------
extern "C" void kernel_launch(void* const* d_in, const int* in_sizes, int n_in,
                              void* d_out, int out_size, void* d_ws, size_t ws_size,
                              hipStream_t stream)
{
    (void)in_sizes; (void)n_in; (void)out_size; (void)ws_size;

    const float* x    = (const float*)d_in[0];   // (32,1,128)
    const float* pk   = (const float*)d_in[1];   // (32,16,2048,128)
    const float* pv   = (const float*)d_in[2];   // (32,16,2048,128)
    const float* Wqkv = (const float*)d_in[3];   // (6144,128)
    const float* bqkv = (const float*)d_in[4];   // (6144,)
    const float* Wout = (const float*)d_in[5];   // (128,2048)
    const float* bout = (const float*)d_in[6];   // (128,)

    float* out  = (float*)d_out;                        // (32,1,128)
    float* outK = out  + (size_t)BB * DD;               // (32,16,2049,128)
    float* outV = outK + (size_t)BB * HH * SP1 * DD;

    float* ws   = (float*)d_ws;
    float* qkv  = ws;                                     // 32*6144
    float* part = qkv  + (size_t)BB * QKVN;               // 512*8*132
    float* y    = part + (size_t)BB * HH * NCHUNK * PSTRIDE;  // 32*2048
    float* z    = y    + (size_t)BB * DT;                 // 32*128

    // 1) QKV projection: (32x128) x (128x6144) via WMMA f32 16x16x4
    wmma_gemm_bias<<<dim3(QKVN / 16, BB / 16), 32, 0, stream>>>(
        x, DD, Wqkv, bqkv, nullptr, qkv, QKVN, DD);

    // 2) Fused KV copy + partial softmax/V accumulation (single NT HBM pass)
    attn_stream<<<dim3(BB * HH * NCHUNK), 256, 0, stream>>>(
        pk, pv, qkv, outK, outV, part);

    // 3) Merge partials + append new K/V row -> y (32x2048)
    attn_reduce<<<dim3(BB * HH), 128, 0, stream>>>(qkv, part, outK, outV, y);

    // 4) Output projection + bias + residual: (32x2048) x (2048x128) via WMMA
    wmma_gemm_bias<<<dim3(DD / 16, BB / 16), 32, 0, stream>>>(
        y, DT, Wout, bout, x, z, DD, DT);

    // 5) LayerNorm -> final out
    layernorm<<<dim3(BB), 128, 0, stream>>>(z, out);
}